// SymmetricChannel_23192823399185
// MI455X (gfx1250) — compile-verified
//
#include <hip/hip_runtime.h>
#include <hip/hip_bf16.h>
#include <stdint.h>

// ---------------------------------------------------------------------------
// SymmetricChannel forward: shifted-mask zeroing of messages + entropy consts.
// Memory-bound streaming kernel (~385 MiB traffic -> ~17us at 23.3 TB/s).
// CDNA5 async Global->LDS (ASYNCcnt) stages each block's noise slab flat in
// LDS (b128 chunks); compute reads the +1-shifted, wrapped window from LDS.
// ---------------------------------------------------------------------------

typedef float v4f __attribute__((ext_vector_type(4)));
typedef int   v4i __attribute__((vector_size(4 * sizeof(int))));

#if defined(__has_builtin)
#if __has_builtin(__builtin_amdgcn_global_load_async_to_lds_b128) && \
    __has_builtin(__builtin_amdgcn_s_wait_asynccnt)
#define USE_ASYNC_B128 1
#endif
#if __has_builtin(__builtin_amdgcn_global_load_async_to_lds_b32) && \
    __has_builtin(__builtin_amdgcn_s_wait_asynccnt)
#define USE_ASYNC_B32 1
#endif
#endif
#ifndef USE_ASYNC_B128
#define USE_ASYNC_B128 0
#endif
#ifndef USE_ASYNC_B32
#define USE_ASYNC_B32 0
#endif

#define ROWS_PER_BLOCK 8
#define THREADS 256
#define N_NOISE (ROWS_PER_BLOCK * 255)   // 2040 floats per block, 16B-aligned slab

// Hb(0.1) + log2(254), computed in double at compile time:
//   -(0.1*log2(0.1) + 0.9*log2(0.9)) = 0.46899559358928117
//   log2(254)                        = 7.988684686772166
static constexpr float K_SYM_CONST = 8.457680280361447f;

typedef __attribute__((address_space(1))) int* gint_p;
typedef __attribute__((address_space(3))) int* lint_p;
typedef __attribute__((address_space(1))) v4i* gv4i_p;
typedef __attribute__((address_space(3))) v4i* lv4i_p;

// ---------------------------------------------------------------------------
// Kernel 1: out[b,l,k] = masked(messages). Column k (k<=253) zeroed iff
// noise[b,l,k+1] < P; column 254 never zeroed; column 255 zeroed iff
// noise[b,l,0] < P. Each block: 8 rows of V=256; noise staged flat in LDS.
// ---------------------------------------------------------------------------
__global__ __launch_bounds__(THREADS) void symchan_mask_kernel(
    const float* __restrict__ messages,
    const float* __restrict__ noise,
    float* __restrict__ out)
{
    __shared__ float lnoise[2048];   // flat: row r at [r*255, r*255+254]; tail pad

    const int    tid  = threadIdx.x;
    const size_t row0 = (size_t)blockIdx.x * ROWS_PER_BLOCK;

    // ---- Stage this block's noise slab (2040 floats, contiguous) into LDS ----
    const float* gsrc = noise + row0 * 255u;   // byte offset blockIdx*8160, 16B-aligned
#if USE_ASYNC_B128
    for (int idx = tid; idx < N_NOISE / 4; idx += THREADS) {   // 510 x b128
        __builtin_amdgcn_global_load_async_to_lds_b128(
            (gv4i_p)((void*)(gsrc + 4 * idx)),
            (lv4i_p)((void*)(&lnoise[4 * idx])),
            0, 0);
    }
    __builtin_amdgcn_s_wait_asynccnt(0);
    __syncthreads();
#elif USE_ASYNC_B32
    for (int idx = tid; idx < N_NOISE; idx += THREADS) {
        __builtin_amdgcn_global_load_async_to_lds_b32(
            (gint_p)((void*)(gsrc + idx)),
            (lint_p)((void*)(&lnoise[idx])),
            0, 0);
    }
    __builtin_amdgcn_s_wait_asynccnt(0);
    __syncthreads();
#else
    for (int idx = tid; idx < N_NOISE; idx += THREADS)
        lnoise[idx] = gsrc[idx];
    __syncthreads();
#endif

    // ---- Compute: wave w handles row w; each lane handles 8 columns ----
    const int    r     = tid >> 5;         // 0..7 (one wave32 per row)
    const int    lane  = tid & 31;
    const int    k0    = lane << 3;        // 8 columns per lane
    const int    nbase = r * 255;          // per-thread constant LDS row base
    const size_t mbase = (row0 + (size_t)r) * 256u;

    const v4f* mrow = (const v4f*)(messages + mbase);
    const v4f  m0   = __builtin_nontemporal_load(mrow + 2 * lane);
    const v4f  m1   = __builtin_nontemporal_load(mrow + 2 * lane + 1);

    v4f o0, o1;
#pragma unroll
    for (int i = 0; i < 8; ++i) {
        const int   k  = k0 + i;
        // k<=253 -> noise col k+1; k==255 -> noise col 0; k==254 -> unused slot
        const float nv = lnoise[nbase + ((k + 1) & 255)];
        const bool  z  = (k != 254) && (nv < 0.1f);
        const float mv = (i < 4) ? m0[i] : m1[i - 4];
        const float ov = z ? 0.0f : mv;
        if (i < 4) o0[i] = ov; else o1[i - 4] = ov;
    }

    __builtin_nontemporal_store(o0, (v4f*)(out + mbase + (size_t)k0));
    __builtin_nontemporal_store(o1, (v4f*)(out + mbase + (size_t)k0 + 4));
}

// ---------------------------------------------------------------------------
// Kernel 2: entropy outputs. One wave32 per batch row b (L = 32 == wave size):
//   sym_ent[b,l]  = entropy[b,l] + C
//   ent_copy[b,l] = entropy[b,l]
//   ent_sum[b]    = sum_l entropy[b,l]
//   sym_sum[b]    = ent_sum[b] + 32*C
// ---------------------------------------------------------------------------
__global__ __launch_bounds__(256) void symchan_entropy_kernel(
    const float* __restrict__ entropy,
    float* __restrict__ sym_sum,    // [B]
    float* __restrict__ sym_ent,    // [B,L]
    float* __restrict__ ent_sum,    // [B]
    float* __restrict__ ent_copy)   // [B,L]
{
    const int lane = threadIdx.x & 31;
    const int wave = threadIdx.x >> 5;
    const int b    = blockIdx.x * 8 + wave;
    const int i    = (b << 5) | lane;

    const float e = entropy[i];
    sym_ent[i]  = e + K_SYM_CONST;
    ent_copy[i] = e;

    float s = e;
#pragma unroll
    for (int off = 16; off > 0; off >>= 1)
        s += __shfl_down(s, off, 32);

    if (lane == 0) {
        ent_sum[b] = s;
        sym_sum[b] = s + 32.0f * K_SYM_CONST;
    }
}

// ---------------------------------------------------------------------------
extern "C" void kernel_launch(void* const* d_in, const int* in_sizes, int n_in,
                              void* d_out, int out_size, void* d_ws, size_t ws_size,
                              hipStream_t stream)
{
    const float* messages = (const float*)d_in[0];  // [B,L,V]   B*L*V floats
    const float* entropy  = (const float*)d_in[1];  // [B,L]     B*L floats
    const float* noise    = (const float*)d_in[2];  // [B,L,V-1] B*L*(V-1) floats

    const int n_msg = in_sizes[0];   // B*L*V = 33,554,432
    const int BL    = in_sizes[1];   // B*L   = 131,072
    const int B     = BL / 32;       // 4,096

    // d_out layout: [out, sym_ent.sum, sym_ent, entropy.sum, entropy]
    float* out      = (float*)d_out;
    float* sym_sum  = out + (size_t)n_msg;
    float* sym_ent  = sym_sum + B;
    float* ent_sum  = sym_ent + BL;
    float* ent_copy = ent_sum + B;

    symchan_mask_kernel<<<dim3(BL / ROWS_PER_BLOCK), THREADS, 0, stream>>>(
        messages, noise, out);

    symchan_entropy_kernel<<<dim3(B / 8), 256, 0, stream>>>(
        entropy, sym_sum, sym_ent, ent_sum, ent_copy);

    (void)n_in; (void)out_size; (void)d_ws; (void)ws_size;
}